// MelGroupConv_35107062678333
// MI455X (gfx1250) — compile-verified
//
#include <hip/hip_runtime.h>

typedef float v2f __attribute__((ext_vector_type(2)));
typedef float v8f __attribute__((ext_vector_type(8)));

#define NGROUPS 16
#define HIDDEN  64
#define SUB     8
#define BATCH   32
#define FREQ    257
#define TLEN    1000
#define TTILE   64
#define XCOLS   82      // x tile: t0-2 .. t0+79 (conv1 halo for the 80-wide h tile)
#define XROWS   48      // max group size padded to multiple of 4
#define HCOLS   80      // h tile: t0-1 .. t0+78 (conv2 halo for the 64-wide y tile)
#define HSTRIDE 84      // row stride: 8*84%64==32, 2*84%64==40 -> conflict-free halves

__constant__ int c_gsize[NGROUPS]  = {6,6,6,6,6,7,8,9,12,14,17,21,25,31,38,45};
__constant__ int c_gstart[NGROUPS] = {0,6,12,18,24,30,37,45,54,66,80,97,118,143,174,212};

struct W1Ptrs { const float* p[NGROUPS]; };

__global__ __launch_bounds__(32)
void melgroupconv_wmma(const float* __restrict__ x,
                       W1Ptrs w1,
                       const float* __restrict__ b1,
                       const float* __restrict__ w2,
                       const float* __restrict__ b2,
                       float* __restrict__ out)
{
    __shared__ float xtile[XROWS * XCOLS];     // 15.7 KB zero-padded input tile
    __shared__ float hbuf[HIDDEN * HSTRIDE];   // 21.5 KB hidden tile

    const int lane  = threadIdx.x;
    const int nloc  = lane & 15;   // M (A rows) / N (B,D cols) position within a 16-tile
    const int khalf = lane >> 4;   // K sub-block {0,1} vs {2,3}
    const int kstride = 2 * khalf;

    const int tile = blockIdx.x;
    const int gi   = blockIdx.y;
    const int b    = blockIdx.z;

    const int g   = c_gsize[gi];
    const int f0  = c_gstart[gi];
    const float* __restrict__ w1p = w1.p[gi];
    const int t0  = tile * TTILE;
    const int gp4 = (g + 3) >> 2;    // K blocks per tap
    const int gp  = gp4 << 2;        // padded channel count

    // ---- Phase 0: stage x into LDS, zero-padded in time and channel ----
    for (int c = 0; c < gp; ++c) {
        const float* __restrict__ xr = x + (size_t)(b * FREQ + f0 + c) * TLEN;
        for (int col = lane; col < XCOLS; col += 32) {
            const int t = t0 - 2 + col;
            float v = 0.f;
            if (c < g && t >= 0 && t < TLEN) v = xr[t];
            xtile[c * XCOLS + col] = v;
        }
    }
    __syncthreads();

    // ---- Phase 1: conv1, per-tap GEMMs, 4 M-tiles x 5 N-tiles held in registers ----
    v8f acc[4][5];
    #pragma unroll
    for (int mt = 0; mt < 4; ++mt)
        #pragma unroll
        for (int nt = 0; nt < 5; ++nt)
            acc[mt][nt] = (v8f){};

    for (int kh = 0; kh < 3; ++kh) {
        const int bcol0 = nloc + kh;            // xtile column base for this tap
        for (int kb = 0; kb < gp4; ++kb) {
            const int c0 = kb * 4 + kstride;
            v2f bf[5], af[4];
            #pragma unroll
            for (int r = 0; r < 2; ++r) {
                const int cr = c0 + r;
                #pragma unroll
                for (int nt = 0; nt < 5; ++nt)
                    bf[nt][r] = xtile[cr * XCOLS + nt * 16 + bcol0];
                const bool cv = (cr < g);
                const int ccl = cv ? cr : 0;    // clamp address, select value
                #pragma unroll
                for (int mt = 0; mt < 4; ++mt) {
                    const float wv = w1p[((mt * 16 + nloc) * g + ccl) * 3 + kh];
                    af[mt][r] = cv ? wv : 0.f;
                }
            }
            #pragma unroll
            for (int mt = 0; mt < 4; ++mt)
                #pragma unroll
                for (int nt = 0; nt < 5; ++nt)
                    acc[mt][nt] = __builtin_amdgcn_wmma_f32_16x16x4_f32(
                        false, af[mt], false, bf[nt], (short)0, acc[mt][nt], false, false);
        }
    }

    // ---- Phase 1b: h tile -> LDS with bias; zero cols outside [0,T) (= conv2 padding) ----
    #pragma unroll
    for (int nt = 0; nt < 5; ++nt) {
        const int col = nt * 16 + nloc;
        const int tg  = t0 - 1 + col;
        const bool tin = (tg >= 0 && tg < TLEN);
        #pragma unroll
        for (int mt = 0; mt < 4; ++mt) {
            #pragma unroll
            for (int rr = 0; rr < 8; ++rr) {
                const int m = mt * 16 + khalf * 8 + rr;
                hbuf[m * HSTRIDE + col] = tin ? (acc[mt][nt][rr] + b1[gi * HIDDEN + m]) : 0.f;
            }
        }
    }
    __syncthreads();

    // ---- Phase 2: conv2 (8->pad16, K=64 per tap), 4 N-tiles in registers ----
    v8f acc2[4];
    #pragma unroll
    for (int yt = 0; yt < 4; ++yt) acc2[yt] = (v8f){};

    const bool svalid = (nloc < SUB);
    const int  srow   = nloc & 7;
    for (int kh = 0; kh < 3; ++kh) {
        const int bcol0 = nloc + kh;
        #pragma unroll 4
        for (int kb = 0; kb < 16; ++kb) {
            const int o0 = kb * 4 + kstride;
            v2f af2, bf2[4];
            #pragma unroll
            for (int r = 0; r < 2; ++r) {
                const int o = o0 + r;
                const float wv = w2[((gi * SUB + srow) * HIDDEN + o) * 3 + kh];
                af2[r] = svalid ? wv : 0.f;
                #pragma unroll
                for (int yt = 0; yt < 4; ++yt)
                    bf2[yt][r] = hbuf[o * HSTRIDE + yt * 16 + bcol0];
            }
            #pragma unroll
            for (int yt = 0; yt < 4; ++yt)
                acc2[yt] = __builtin_amdgcn_wmma_f32_16x16x4_f32(
                    false, af2, false, bf2[yt], (short)0, acc2[yt], false, false);
        }
    }

    // ---- Phase 2b: write y; D rows 0..7 live in lanes 0..15, vgpr index = subband ----
    #pragma unroll
    for (int yt = 0; yt < 4; ++yt) {
        const int t = t0 + yt * 16 + nloc;
        if (khalf == 0 && t < TLEN) {
            #pragma unroll
            for (int rr = 0; rr < SUB; ++rr) {
                out[(((size_t)(b * NGROUPS + gi)) * SUB + rr) * TLEN + t] =
                    acc2[yt][rr] + b2[gi * SUB + rr];
            }
        }
    }
}

extern "C" void kernel_launch(void* const* d_in, const int* in_sizes, int n_in,
                              void* d_out, int out_size, void* d_ws, size_t ws_size,
                              hipStream_t stream) {
    (void)in_sizes; (void)n_in; (void)out_size; (void)d_ws; (void)ws_size;
    const float* x = (const float*)d_in[0];
    W1Ptrs w1;
    for (int i = 0; i < NGROUPS; ++i) w1.p[i] = (const float*)d_in[1 + i];
    const float* b1 = (const float*)d_in[17];
    const float* w2 = (const float*)d_in[18];
    const float* b2 = (const float*)d_in[19];
    float* out = (float*)d_out;

    dim3 grid((TLEN + TTILE - 1) / TTILE, NGROUPS, BATCH);   // 16 x 16 x 32 waves
    melgroupconv_wmma<<<grid, 32, 0, stream>>>(x, w1, b1, w2, b2, out);
}